// DeepERA_9174050144549
// MI455X (gfx1250) — compile-verified
//
#include <hip/hip_runtime.h>
#include <math.h>

#define DIM 64
#define NC 2000
#define NP 1500
#define NATOMS 40
#define LPROT 512
#define LGNN 3
#define LCNN 3
#define LGCN 2
#define WIN 5
#define BATCH 4096
#define KW 11            // 2*WIN+1
#define KTAPS (KW*KW)    // 121
#define KT 64            // GEMM k-chunk staged in LDS
#define BPAD 2           // LDS row padding (bank-conflict kill)

typedef float v2f __attribute__((ext_vector_type(2)));
typedef float v8f __attribute__((ext_vector_type(8)));

// ---------------------------------------------------------------------------
// f32 WMMA: D(16x16) = A(16x4) * B(4x16) + C   (CDNA5 V_WMMA_F32_16X16X4_F32)
// ---------------------------------------------------------------------------
__device__ __forceinline__ v8f wmma_f32x4(v2f a, v2f b, v8f c) {
  return __builtin_amdgcn_wmma_f32_16x16x4_f32(
      /*neg_a=*/false, a, /*neg_b=*/false, b,
      /*c_mod=*/(short)0, c, /*reuse_a=*/false, /*reuse_b=*/false);
}

// Async global->LDS staging (CDNA5, ASYNCcnt-tracked, no VGPR landing zone)
__device__ __forceinline__ void async_load_f32_to_lds(float* ldsDst,
                                                      const float* gSrc) {
  unsigned lAddr = (unsigned)(size_t)ldsDst;                 // LDS offset
  unsigned long long gAddr = (unsigned long long)(size_t)gSrc;
  asm volatile("global_load_async_to_lds_b32 %0, %1, off"
               :: "v"(lAddr), "v"(gAddr) : "memory");
}
__device__ __forceinline__ void wait_asynccnt0() {
  asm volatile("s_wait_asynccnt 0x0" ::: "memory");
}

// ---------------------------------------------------------------------------
// WMMA GEMM:  C[M,N] = act( A[g(m),0:K] @ B[0:K,0:N] + bias[n] )
//   Requires K % 4 == 0 (true for every GEMM in this net).
//   Grid: x = M-supertiles (8 waves x 16 rows), y = 64-wide N-groups.
//   B panel (KT x 64) double-buffered in LDS: async DMA of chunk i+1 overlaps
//   the 16 WMMA k-steps of chunk i.   act: 0=none 1=relu 2=sigmoid
// ---------------------------------------------------------------------------
__global__ void __launch_bounds__(256)
gemm_wmma_kernel(const float* __restrict__ A, int lda,
                 const int* __restrict__ idxA,
                 const float* __restrict__ B, int ldb,
                 const float* __restrict__ bias,
                 float* __restrict__ C, int ldc,
                 int M, int N, int K, int act)
{
  __shared__ float Blds[2][KT][DIM + BPAD];        // ~34 KB, ping-pong
  const int tid  = threadIdx.x;
  const int lane = tid & 31;
  const int hlf  = lane >> 4;                      // 0: K0/K1, 1: K2/K3
  const int l    = lane & 15;
  const int wv   = tid >> 5;                       // 0..7
  const int mbase = ((int)blockIdx.x * 8 + wv) * 16;
  const int nbase = (int)blockIdx.y * 64;
  const bool fullN = (nbase + 64) <= N;

  const int  row   = mbase + l;
  const bool rowOK = row < M;
  const int  rowC  = rowOK ? row : (M - 1);        // clamp -> branchless loads
  const long arow  = (long)(idxA ? idxA[rowC] : rowC) * (long)lda;

  v8f acc[4];
  const v8f zero = {0.f,0.f,0.f,0.f,0.f,0.f,0.f,0.f};
#pragma unroll
  for (int t = 0; t < 4; ++t) acc[t] = zero;

  const int nChunks = (K + KT - 1) / KT;

  // stage B[chunk*KT + kk][nbase+c] -> Blds[buf][kk][c]
  auto stage = [&](int chunk, int buf) {
    const int k0 = chunk * KT;
    const int kcount = (K - k0) < KT ? (K - k0) : KT;   // multiple of 4
#pragma unroll 1
    for (int idx = tid; idx < KT * DIM; idx += 256) {
      const int kk = idx >> 6, c = idx & 63;
      if (fullN && kk < kcount) {
        async_load_f32_to_lds(&Blds[buf][kk][c],
                              B + (long)(k0 + kk) * ldb + nbase + c);
      } else {
        const int c2 = nbase + c;
        Blds[buf][kk][c] = (kk < kcount && c2 < N)
                             ? B[(long)(k0 + kk) * ldb + c2] : 0.0f;
      }
    }
  };

  stage(0, 0);
  wait_asynccnt0();
  __syncthreads();

  for (int chunk = 0; chunk < nChunks; ++chunk) {
    const int buf = chunk & 1;
    // kick next chunk's DMA; it drains behind this chunk's WMMA work
    if (chunk + 1 < nChunks) {
      stage(chunk + 1, buf ^ 1);
      __builtin_prefetch(A + arow + (chunk + 1) * KT + lane * 2, 0, 1);
    }

    const int k0 = chunk * KT;
    const int kcount = (K - k0) < KT ? (K - k0) : KT;
    const int ksteps = kcount >> 2;
#pragma unroll 4
    for (int ks = 0; ks < ksteps; ++ks) {
      const int ka = ks * 4 + hlf * 2;
      const v2f a = *(const v2f*)(A + arow + k0 + ka);   // global_load_b64
#pragma unroll
      for (int t = 0; t < 4; ++t) {
        v2f b;
        b.x = Blds[buf][ka    ][t * 16 + l];
        b.y = Blds[buf][ka + 1][t * 16 + l];
        acc[t] = wmma_f32x4(a, b, acc[t]);
      }
    }
    wait_asynccnt0();     // own DMA for chunk+1 done
    __syncthreads();      // everyone finished reading buf & writing buf^1
  }

  // ---- epilogue: bias + activation, guarded stores ----
#pragma unroll
  for (int t = 0; t < 4; ++t) {
    const int col = nbase + t * 16 + l;
    if (col >= N) continue;
    const float bv = bias ? bias[col] : 0.0f;
#pragma unroll
    for (int v = 0; v < 8; ++v) {
      const int r = mbase + v + 8 * hlf;
      if (r >= M) continue;
      float val = acc[t][v] + bv;
      if (act == 1)      val = fmaxf(val, 0.0f);
      else if (act == 2) val = 1.0f / (1.0f + expf(-val));
      C[(long)r * ldc + col] = val;
    }
  }
}

// ---------------------------------------------------------------------------
// GNN residual step: xs[n] += adj[n](40x40) @ hs[n](40x64); 3 waves/compound
// ---------------------------------------------------------------------------
__global__ void __launch_bounds__(96)
gnn_adj_residual_kernel(const float* __restrict__ adj,
                        const float* __restrict__ hs,
                        float* __restrict__ xs)
{
  const int n    = blockIdx.x;
  const int lane = threadIdx.x & 31;
  const int hlf  = lane >> 4;
  const int l    = lane & 15;
  const int mt   = threadIdx.x >> 5;        // 0..2 (rows 0-15,16-31,32-39pad)
  const int mbase = mt * 16;

  const float* Aadj = adj + (long)n * NATOMS * NATOMS;
  const float* Bh   = hs  + (long)n * NATOMS * DIM;
  float*       X    = xs  + (long)n * NATOMS * DIM;

  const int row  = mbase + l;
  const int rowC = row < NATOMS ? row : (NATOMS - 1);   // branchless clamp

  v8f acc[4];
  const v8f zero = {0.f,0.f,0.f,0.f,0.f,0.f,0.f,0.f};
#pragma unroll
  for (int t = 0; t < 4; ++t) acc[t] = zero;

#pragma unroll 2
  for (int k0 = 0; k0 < NATOMS; k0 += 4) {   // K = 40
    const int ka = k0 + hlf * 2;
    const v2f a = *(const v2f*)(Aadj + rowC * NATOMS + ka);
#pragma unroll
    for (int t = 0; t < 4; ++t) {
      const int col = t * 16 + l;
      v2f b;
      b.x = Bh[(ka    ) * DIM + col];
      b.y = Bh[(ka + 1) * DIM + col];
      acc[t] = wmma_f32x4(a, b, acc[t]);
    }
  }
#pragma unroll
  for (int t = 0; t < 4; ++t) {
    const int col = t * 16 + l;
#pragma unroll
    for (int v = 0; v < 8; ++v) {
      const int r = mbase + v + 8 * hlf;
      if (r < NATOMS) X[r * DIM + col] += acc[t][v];
    }
  }
}

// ---------------------------------------------------------------------------
__global__ void gather_fp_kernel(const int* __restrict__ compounds,
                                 const float* __restrict__ embed_fp,
                                 float* __restrict__ xs)
{
  const int i = blockIdx.x * blockDim.x + threadIdx.x;
  if (i >= NC * NATOMS * DIM) return;
  const int row = i >> 6;
  const int d   = i & 63;
  xs[i] = embed_fp[(long)compounds[row] * DIM + d];
}

__global__ void mean_atoms_kernel(const float* __restrict__ xs,
                                  float* __restrict__ comp_int)
{
  const int n = blockIdx.x;
  const int d = threadIdx.x;   // 64 threads
  float s = 0.0f;
  for (int a = 0; a < NATOMS; ++a) s += xs[((long)n * NATOMS + a) * DIM + d];
  comp_int[(long)n * DIM + d] = s * (1.0f / NATOMS);
}

// ---------------------------------------------------------------------------
// Protein CNN: one workgroup per protein; the 512x64 activation plane
// ping-pongs entirely inside LDS (2 x 128 KB of the WGP's 320 KB), fused mean.
// ---------------------------------------------------------------------------
__global__ void __launch_bounds__(256)
protein_cnn_kernel(const int* __restrict__ proteins,
                   const float* __restrict__ embed_word,
                   const float* __restrict__ K_cnn,
                   const float* __restrict__ b_cnn,
                   float* __restrict__ prot_int)
{
  extern __shared__ float lds[];
  float* buf0 = lds;
  float* buf1 = lds + LPROT * DIM;
  float* kern = lds + 2 * LPROT * DIM;   // 121 taps
  const int p   = blockIdx.x;
  const int tid = threadIdx.x;

  for (int idx = tid; idx < LPROT * DIM; idx += 256) {
    const int ll = idx >> 6, d = idx & 63;
    const int w  = proteins[p * LPROT + ll];
    buf0[idx] = embed_word[(long)w * DIM + d];
  }
  __syncthreads();

  float* in  = buf0;
  float* out = buf1;
  for (int layer = 0; layer < LCNN; ++layer) {
    if (tid < KTAPS) kern[tid] = K_cnn[layer * KTAPS + tid];
    __syncthreads();
    const float bias = b_cnn[layer];
    for (int idx = tid; idx < LPROT * DIM; idx += 256) {
      const int ll = idx >> 6, d = idx & 63;
      float acc = bias;
#pragma unroll
      for (int dl = -WIN; dl <= WIN; ++dl) {
        const int l2 = ll + dl;
        if ((unsigned)l2 >= (unsigned)LPROT) continue;
        const float* rowp = in + l2 * DIM;
        const float* kp   = kern + (dl + WIN) * KW;
#pragma unroll
        for (int dd = -WIN; dd <= WIN; ++dd) {
          const int d2 = d + dd;
          if ((unsigned)d2 < (unsigned)DIM) acc += kp[dd + WIN] * rowp[d2];
        }
      }
      out[idx] = fmaxf(acc, 0.0f);
    }
    __syncthreads();
    float* t = in; in = out; out = t;
  }
  if (tid < DIM) {
    float s = 0.0f;
    for (int ll = 0; ll < LPROT; ++ll) s += in[ll * DIM + tid];
    prot_int[(long)p * DIM + tid] = s * (1.0f / LPROT);
  }
}

// ---------------------------------------------------------------------------
// concat [B, 6*DIM] : comp_int[ic] | Xc[ic] | fd | prot_int[ip] | Xp[ip] | fp
// ---------------------------------------------------------------------------
__global__ void concat_kernel(const int* __restrict__ idx_c,
                              const int* __restrict__ idx_p,
                              const float* __restrict__ comp_int,
                              const float* __restrict__ Xc,
                              const float* __restrict__ fd,
                              const float* __restrict__ prot_int,
                              const float* __restrict__ Xp,
                              const float* __restrict__ fp,
                              float* __restrict__ cat)
{
  const int i = blockIdx.x * blockDim.x + threadIdx.x;
  if (i >= BATCH * 6 * DIM) return;
  const int r   = i / (6 * DIM);
  const int c   = i % (6 * DIM);
  const int seg = c >> 6;
  const int d   = c & 63;
  float v;
  switch (seg) {
    case 0:  v = comp_int[(long)idx_c[r] * DIM + d]; break;
    case 1:  v = Xc      [(long)idx_c[r] * DIM + d]; break;
    case 2:  v = fd      [(long)r        * DIM + d]; break;
    case 3:  v = prot_int[(long)idx_p[r] * DIM + d]; break;
    case 4:  v = Xp      [(long)idx_p[r] * DIM + d]; break;
    default: v = fp      [(long)r        * DIM + d]; break;
  }
  cat[i] = v;
}

// ---------------------------------------------------------------------------
static inline void launch_gemm(const float* A, int lda, const int* idxA,
                               const float* B, int ldb, const float* bias,
                               float* C, int ldc, int M, int N, int K, int act,
                               hipStream_t s)
{
  dim3 grid((unsigned)(((M + 15) / 16 + 7) / 8), (unsigned)((N + 63) / 64));
  gemm_wmma_kernel<<<grid, 256, 0, s>>>(A, lda, idxA, B, ldb, bias,
                                        C, ldc, M, N, K, act);
}

extern "C" void kernel_launch(void* const* d_in, const int* in_sizes, int n_in,
                              void* d_out, int out_size, void* d_ws, size_t ws_size,
                              hipStream_t stream)
{
  (void)in_sizes; (void)n_in; (void)out_size; (void)ws_size;
  const int*   idx_c        = (const int*)  d_in[0];
  const int*   idx_p        = (const int*)  d_in[1];
  const int*   compounds    = (const int*)  d_in[2];
  const int*   proteins     = (const int*)  d_in[3];
  const float* adjacencies  = (const float*)d_in[4];
  const float* A_c          = (const float*)d_in[5];
  const float* A_p          = (const float*)d_in[6];
  const float* embed_fp     = (const float*)d_in[7];
  const float* embed_word   = (const float*)d_in[8];
  const float* Xs_c         = (const float*)d_in[9];
  const float* Xs_p         = (const float*)d_in[10];
  const float* drug_feat    = (const float*)d_in[11];
  const float* protein_feat = (const float*)d_in[12];
  const float* W_gnn        = (const float*)d_in[13];
  const float* b_gnn        = (const float*)d_in[14];
  const float* K_cnn        = (const float*)d_in[15];
  const float* b_cnn        = (const float*)d_in[16];
  const float* W_gcn_d      = (const float*)d_in[17];
  const float* b_gcn_d      = (const float*)d_in[18];
  const float* W_gcn_p      = (const float*)d_in[19];
  const float* b_gcn_p      = (const float*)d_in[20];
  const float* Wd1          = (const float*)d_in[21];
  const float* bd1          = (const float*)d_in[22];
  const float* Wd2          = (const float*)d_in[23];
  const float* bd2          = (const float*)d_in[24];
  const float* Wd3          = (const float*)d_in[25];
  const float* bd3          = (const float*)d_in[26];
  const float* Wp1          = (const float*)d_in[27];
  const float* bp1          = (const float*)d_in[28];
  const float* Wp2          = (const float*)d_in[29];
  const float* bp2          = (const float*)d_in[30];
  const float* Wp3          = (const float*)d_in[31];
  const float* bp3          = (const float*)d_in[32];
  const float* Wo1          = (const float*)d_in[33];
  const float* bo1          = (const float*)d_in[34];
  const float* Wo2          = (const float*)d_in[35];
  const float* bo2          = (const float*)d_in[36];
  const float* Wo3          = (const float*)d_in[37];
  const float* bo3          = (const float*)d_in[38];
  const float* W_int        = (const float*)d_in[39];
  const float* b_int        = (const float*)d_in[40];
  float* out = (float*)d_out;

  // ---- workspace layout (f32) ----
  float* ws = (float*)d_ws;
  size_t o = 0;
  float* xs       = ws + o; o += (size_t)NC * NATOMS * DIM;
  float* hs       = ws + o; o += (size_t)NC * NATOMS * DIM;
  float* comp_int = ws + o; o += (size_t)NC * DIM;
  float* prot_int = ws + o; o += (size_t)NP * DIM;
  float* Tc       = ws + o; o += (size_t)NC * DIM;
  float* Xc_a     = ws + o; o += (size_t)NC * DIM;
  float* Xc_b     = ws + o; o += (size_t)NC * DIM;
  float* Tp       = ws + o; o += (size_t)NP * DIM;
  float* Xp_a     = ws + o; o += (size_t)NP * DIM;
  float* Xp_b     = ws + o; o += (size_t)NP * DIM;
  float* fd1      = ws + o; o += (size_t)BATCH * 2 * DIM;
  float* fd2      = ws + o; o += (size_t)BATCH * DIM;
  float* fd3      = ws + o; o += (size_t)BATCH * DIM;
  float* fp1      = ws + o; o += (size_t)BATCH * 2 * DIM;
  float* fp2      = ws + o; o += (size_t)BATCH * DIM;
  float* fp3      = ws + o; o += (size_t)BATCH * DIM;
  float* cat      = ws + o; o += (size_t)BATCH * 6 * DIM;
  float* h1       = ws + o; o += (size_t)BATCH * 4 * DIM;
  float* h2       = ws + o; o += (size_t)BATCH * 4 * DIM;
  float* h3       = ws + o; o += (size_t)BATCH * 4 * DIM;

  // ---- Stage 1: compound GNN ----
  {
    const int n = NC * NATOMS * DIM;
    gather_fp_kernel<<<(n + 255) / 256, 256, 0, stream>>>(compounds, embed_fp, xs);
  }
  for (int i = 0; i < LGNN; ++i) {
    launch_gemm(xs, DIM, nullptr, W_gnn + (size_t)i * DIM * DIM, DIM,
                b_gnn + (size_t)i * DIM, hs, DIM,
                NC * NATOMS, DIM, DIM, /*relu*/1, stream);
    gnn_adj_residual_kernel<<<NC, 96, 0, stream>>>(adjacencies, hs, xs);
  }
  mean_atoms_kernel<<<NC, DIM, 0, stream>>>(xs, comp_int);

  // ---- Stage 2: protein CNN (LDS-resident plane, fused mean) ----
  {
    const size_t ldsBytes = (size_t)(2 * LPROT * DIM + 128) * sizeof(float);
    protein_cnn_kernel<<<NP, 256, ldsBytes, stream>>>(proteins, embed_word,
                                                      K_cnn, b_cnn, prot_int);
  }

  // ---- Stage 3: GCN diffusion ----
  {
    const float* Xcur = Xs_c;
    float* bufs[2] = {Xc_a, Xc_b};
    for (int i = 0; i < LGCN; ++i) {
      launch_gemm(A_c, NC, nullptr, Xcur, DIM, nullptr, Tc, DIM,
                  NC, DIM, NC, /*none*/0, stream);
      launch_gemm(Tc, DIM, nullptr, W_gcn_d + (size_t)i * DIM * DIM, DIM,
                  b_gcn_d + (size_t)i * DIM, bufs[i], DIM,
                  NC, DIM, DIM, /*relu*/1, stream);
      Xcur = bufs[i];
    }
  }
  {
    const float* Xcur = Xs_p;
    float* bufs[2] = {Xp_a, Xp_b};
    for (int i = 0; i < LGCN; ++i) {
      launch_gemm(A_p, NP, nullptr, Xcur, DIM, nullptr, Tp, DIM,
                  NP, DIM, NP, /*none*/0, stream);
      launch_gemm(Tp, DIM, nullptr, W_gcn_p + (size_t)i * DIM * DIM, DIM,
                  b_gcn_p + (size_t)i * DIM, bufs[i], DIM,
                  NP, DIM, DIM, /*relu*/1, stream);
      Xcur = bufs[i];
    }
  }

  // ---- Stage 4: annotation MLPs (gather fused into first GEMM) ----
  launch_gemm(drug_feat, 1024, idx_c, Wd1, 2 * DIM, bd1, fd1, 2 * DIM,
              BATCH, 2 * DIM, 1024, 1, stream);
  launch_gemm(fd1, 2 * DIM, nullptr, Wd2, DIM, bd2, fd2, DIM,
              BATCH, DIM, 2 * DIM, 1, stream);
  launch_gemm(fd2, DIM, nullptr, Wd3, DIM, bd3, fd3, DIM,
              BATCH, DIM, DIM, 1, stream);
  launch_gemm(protein_feat, 1024, idx_p, Wp1, 2 * DIM, bp1, fp1, 2 * DIM,
              BATCH, 2 * DIM, 1024, 1, stream);
  launch_gemm(fp1, 2 * DIM, nullptr, Wp2, DIM, bp2, fp2, DIM,
              BATCH, DIM, 2 * DIM, 1, stream);
  launch_gemm(fp2, DIM, nullptr, Wp3, DIM, bp3, fp3, DIM,
              BATCH, DIM, DIM, 1, stream);

  // ---- Stage 5: concat + output MLP + sigmoid head ----
  {
    const int n = BATCH * 6 * DIM;
    concat_kernel<<<(n + 255) / 256, 256, 0, stream>>>(
        idx_c, idx_p, comp_int, Xc_b, fd3, prot_int, Xp_b, fp3, cat);
  }
  launch_gemm(cat, 6 * DIM, nullptr, Wo1, 4 * DIM, bo1, h1, 4 * DIM,
              BATCH, 4 * DIM, 6 * DIM, 1, stream);
  launch_gemm(h1, 4 * DIM, nullptr, Wo2, 4 * DIM, bo2, h2, 4 * DIM,
              BATCH, 4 * DIM, 4 * DIM, 1, stream);
  launch_gemm(h2, 4 * DIM, nullptr, Wo3, 4 * DIM, bo3, h3, 4 * DIM,
              BATCH, 4 * DIM, 4 * DIM, 1, stream);
  launch_gemm(h3, 4 * DIM, nullptr, W_int, 2, b_int, out, 2,
              BATCH, 2, 4 * DIM, /*sigmoid*/2, stream);
}